// AdaptiveStdPool2d_48679159333426
// MI455X (gfx1250) — compile-verified
//
#include <hip/hip_runtime.h>
#include <hip/hip_bf16.h>

// AdaptiveStdPool2d: x[32,64,512,80] f32 -> out[32,64,8,10] f32
// window = 128x8, mean/std interleaved along height.
// Pure bandwidth-bound streaming reduction (~335 MB @ 23.3 TB/s ~= 14.4 us floor).
//
// Design:
//  - 1 block per (b,c) plane (2048 blocks, 160 threads = 5 full wave32s)
//  - async global->LDS B128 copies, double-buffered 64-row half-slabs (ASYNCcnt pipeline)
//  - column sums via V_WMMA_F32_16X16X4_F32 with A == ones (D[m][n] = sum_k B[k][n]);
//    second WMMA chain on squared elements gives sum(x^2). Layout-robust since A==1.
//  - tiny finalize folds 8 columns per window, computes mean / sqrt(var+eps).

#define NH 512
#define NW 80
#define OH2 4
#define NOW 10
#define WW 8
#define WH 128

#define THREADS 160
#define HALF_ROWS 64
#define HALF_FLOATS (HALF_ROWS * NW)   // 5120 floats
#define HALF_BYTES  (HALF_FLOATS * 4)  // 20480 bytes
#define CHUNKS ((HALF_BYTES / 16) / THREADS)  // 8 x b128 per thread per stage

typedef float v2f __attribute__((ext_vector_type(2)));
typedef float v8f __attribute__((ext_vector_type(8)));
typedef int   v4i __attribute__((__vector_size__(4 * sizeof(int))));
typedef __attribute__((address_space(1))) v4i gv4i;   // global 16B chunk
typedef __attribute__((address_space(3))) v4i lv4i;   // LDS 16B chunk

__global__ __launch_bounds__(THREADS)
void AdaptiveStdPool2d_48679159333426_kernel(const float* __restrict__ x,
                                             float* __restrict__ out)
{
    // two 20KB half-slab buffers + 80 col-sums + 80 col-sumsq  (41.6 KB)
    __shared__ __align__(16) float lds[2 * HALF_FLOATS + 2 * NW];
    float* colS = &lds[2 * HALF_FLOATS];
    float* colQ = colS + NW;

    const int t    = threadIdx.x;
    const int lane = t & 31;
    const int wv   = t >> 5;          // 0..4, owns columns [16*wv, 16*wv+16)
    const int bc   = blockIdx.x;      // 0..2047

    const float*   plane = x + (size_t)bc * (NH * NW);
    const unsigned lds0  = (unsigned)(size_t)&lds[0];

    // issue one 20KB half-slab (stage s = rows [s*64, s*64+64)) into buffer `buf`
    auto issue_stage = [&](int s, int buf) {
        const uintptr_t g0 = (uintptr_t)(plane + (size_t)s * HALF_FLOATS);
        const unsigned  l0 = lds0 + (unsigned)buf * HALF_BYTES;
#pragma unroll
        for (int i = 0; i < CHUNKS; ++i) {
            const unsigned byte = ((unsigned)t + (unsigned)i * THREADS) * 16u;
#if __has_builtin(__builtin_amdgcn_global_load_async_to_lds_b128)
            __builtin_amdgcn_global_load_async_to_lds_b128(
                (gv4i*)(g0 + byte), (lv4i*)(l0 + byte), 0, 0);
#else
            asm volatile("global_load_async_to_lds_b128 %0, %1, off"
                         :: "v"((lv4i*)(l0 + byte)), "v"((gv4i*)(g0 + byte))
                         : "memory");
#endif
        }
    };

    issue_stage(0, 0);

    const int col0 = wv * 16 + (lane & 15);  // column this lane feeds into WMMA-B
    const int rsel = lane >> 4;              // lanes 16..31 carry odd rows
    const v2f ones = {1.0f, 1.0f};

    v8f accS = {0.f, 0.f, 0.f, 0.f, 0.f, 0.f, 0.f, 0.f};
    v8f accQ = {0.f, 0.f, 0.f, 0.f, 0.f, 0.f, 0.f, 0.f};

    for (int s = 0; s < 2 * OH2; ++s) {
        if (s < 2 * OH2 - 1) {
            issue_stage(s + 1, (s + 1) & 1);
            asm volatile("s_wait_asynccnt 8" ::: "memory");  // stage s landed
        } else {
            asm volatile("s_wait_asynccnt 0" ::: "memory");
        }
        __syncthreads();

        const float* buf = &lds[(s & 1) * HALF_FLOATS];

        // 64 rows: 16 chained WMMA steps, each reduces 4 rows x 16 cols
#pragma unroll 8
        for (int rb = 0; rb < HALF_ROWS; rb += 4) {
            const float f0 = buf[(rb + rsel) * NW + col0];
            const float f1 = buf[(rb + 2 + rsel) * NW + col0];
            v2f bS = {f0, f1};
            v2f bQ = {f0 * f0, f1 * f1};
            accS = __builtin_amdgcn_wmma_f32_16x16x4_f32(
                false, ones, false, bS, (short)0, accS, false, false);
            accQ = __builtin_amdgcn_wmma_f32_16x16x4_f32(
                false, ones, false, bQ, (short)0, accQ, false, false);
        }

        if (s & 1) {  // finished 128 rows of window-row oh = s>>1
            const int oh = s >> 1;
            if (lane < 16) {
                // D row M=0: VGPR0, lanes 0..15 hold per-column sums
                colS[col0] = accS[0];
                colQ[col0] = accQ[0];
            }
            __syncthreads();

            if (t < NOW) {
                float sm = 0.f, sq = 0.f;
#pragma unroll
                for (int j = 0; j < WW; ++j) {
                    sm += colS[t * WW + j];
                    sq += colQ[t * WW + j];
                }
                const float inv  = 1.0f / (float)(WH * WW);  // 1/1024
                const float mean = sm * inv;
                const float var  = fmaxf(sq * inv - mean * mean, 0.0f);
                const float sd   = sqrtf(var + 1e-14f);
                float* o = out + (size_t)bc * (2 * OH2 * NOW) + (size_t)(2 * oh) * NOW + t;
                o[0]   = mean;
                o[NOW] = sd;
            }
            accS = (v8f){0.f, 0.f, 0.f, 0.f, 0.f, 0.f, 0.f, 0.f};
            accQ = (v8f){0.f, 0.f, 0.f, 0.f, 0.f, 0.f, 0.f, 0.f};
        } else {
            __syncthreads();  // release buffer (s&1) readers before it is re-filled
        }
    }
}

extern "C" void kernel_launch(void* const* d_in, const int* in_sizes, int n_in,
                              void* d_out, int out_size, void* d_ws, size_t ws_size,
                              hipStream_t stream) {
    (void)in_sizes; (void)n_in; (void)d_ws; (void)ws_size; (void)out_size;
    const float* x = (const float*)d_in[0];
    float* out     = (float*)d_out;
    dim3 grid(32 * 64);     // one block per (b,c) plane
    dim3 block(THREADS);    // 5 wave32s
    hipLaunchKernelGGL(AdaptiveStdPool2d_48679159333426_kernel,
                       grid, block, 0, stream, x, out);
}